// Mamba_46797963657514
// MI455X (gfx1250) — compile-verified
//
#include <hip/hip_runtime.h>
#include <hip/hip_bf16.h>
#include <math.h>

// ---------------- problem constants (must match reference) ----------------
#define B_   8
#define L_   2048
#define DM   128
#define NL   4
#define DI   256
#define DC   8
#define DS   16
#define DTR  8
#define NTOK (B_ * L_)          // 16384 rows
#define DXZ  (2 * DI)           // 512
#define NDBC (DTR + 2 * DS)     // 40

typedef __attribute__((ext_vector_type(2))) float v2f;
typedef __attribute__((ext_vector_type(8))) float v8f;

__device__ __forceinline__ float sigmoidf_(float x) {
    return 1.0f / (1.0f + __expf(-x));
}

// ---------------------------------------------------------------------------
// 1) RMSNorm over DM=128: one block (128 threads = 4 waves) per row.
// ---------------------------------------------------------------------------
__global__ void rmsnorm_kernel(const float* __restrict__ x,
                               const float* __restrict__ w,
                               float* __restrict__ xn) {
    __shared__ float red[4];
    const int row = blockIdx.x;
    const int tid = threadIdx.x;            // 0..127
    float v = x[row * DM + tid];
    float s = v * v;
    #pragma unroll
    for (int off = 16; off > 0; off >>= 1)
        s += __shfl_xor(s, off, 32);        // wave32 reduction
    if ((tid & 31) == 0) red[tid >> 5] = s;
    __syncthreads();
    float tot = red[0] + red[1] + red[2] + red[3];
    float r = rsqrtf(tot * (1.0f / DM) + 1e-5f);
    xn[row * DM + tid] = v * r * w[tid];
}

// ---------------------------------------------------------------------------
// 2) in_proj: xz = xn @ Wi^T     Wi:(2*DI, DM) row-major, K = DM = 128.
//    WMMA f32 16x16x4. Each wave keeps its full-K A fragment (32 x v2f =
//    64 VGPRs) in registers and sweeps 8 N tiles -> 8x A reuse.
//    grid = NTOK/16 blocks, 128 threads (4 waves).
// ---------------------------------------------------------------------------
__global__ void inproj_wmma_kernel(const float* __restrict__ xn,
                                   const float* __restrict__ W,
                                   float* __restrict__ xz) {
    const int lane = threadIdx.x & 31;
    const int wave = threadIdx.x >> 5;   // 0..3
    const int m0 = blockIdx.x * 16;
    const int t  = lane & 15;       // A: row in tile / B: col in tile
    const int hi = lane >> 4;       // half-wave select
    const int ko = hi * 2;          // K sub-offset 0 or 2 (ISA f32 A layout)

    // A fragment for the entire K=128 (resident in VGPRs)
    v2f afrag[DM / 4];
    {
        const float* __restrict__ arow = xn + (size_t)(m0 + t) * DM;
        #pragma unroll
        for (int kk = 0; kk < DM / 4; ++kk) {
            afrag[kk][0] = arow[kk * 4 + ko];
            afrag[kk][1] = arow[kk * 4 + ko + 1];
        }
    }

    // wave handles n-tiles wave, wave+4, ..., wave+28  (uniform per wave,
    // EXEC stays all-ones around the WMMAs)
    for (int nt = wave; nt < DXZ / 16; nt += 4) {
        const int n0 = nt * 16;
        const float* __restrict__ brow = W + (size_t)(n0 + t) * DM;
        v8f acc = {};
        #pragma unroll
        for (int kk = 0; kk < DM / 4; ++kk) {
            v2f b;
            b[0] = brow[kk * 4 + ko];
            b[1] = brow[kk * 4 + ko + 1];
            acc = __builtin_amdgcn_wmma_f32_16x16x4_f32(
                false, afrag[kk], false, b, (short)0, acc, false, false);
        }
        // C/D layout: vgpr r -> row m0 + hi*8 + r, col n0 + t
        float* __restrict__ ocol = xz + n0 + t;
        #pragma unroll
        for (int r = 0; r < 8; ++r)
            ocol[(size_t)(m0 + hi * 8 + r) * DXZ] = acc[r];
    }
}

// ---------------------------------------------------------------------------
// 3) depthwise causal conv (DC=8) + bias + SiLU on xf = xz[:, :DI]
// ---------------------------------------------------------------------------
__global__ void conv_silu_kernel(const float* __restrict__ xz,
                                 const float* __restrict__ cw,  // (DI, DC)
                                 const float* __restrict__ cb,  // (DI)
                                 float* __restrict__ xc) {
    const int idx = blockIdx.x * 256 + threadIdx.x;   // over NTOK*DI
    const int di  = idx & (DI - 1);
    const int row = idx / DI;
    const int l   = row & (L_ - 1);
    const int b   = row / L_;
    float acc = cb[di];
    #pragma unroll
    for (int j = 0; j < DC; ++j) {
        int lj = l - (DC - 1) + j;
        if (lj >= 0)
            acc += cw[di * DC + j] * xz[((size_t)(b * L_ + lj)) * DXZ + di];
    }
    xc[idx] = acc * sigmoidf_(acc);   // SiLU
}

// ---------------------------------------------------------------------------
// 4) x_proj (40 outputs, K=256) + dt_proj (K=8) + softplus, one block/row.
// ---------------------------------------------------------------------------
__global__ void xproj_kernel(const float* __restrict__ xc,
                             const float* __restrict__ xw,  // (40, DI)
                             const float* __restrict__ dw,  // (DI, DTR)
                             const float* __restrict__ db,  // (DI)
                             float* __restrict__ dlt,       // (NTOK, DI)
                             float* __restrict__ Bm,        // (NTOK, DS)
                             float* __restrict__ Cm) {      // (NTOK, DS)
    __shared__ float sx[DI];
    __shared__ float sd[NDBC];
    const int row = blockIdx.x;
    const int tid = threadIdx.x;          // 256
    sx[tid] = xc[(size_t)row * DI + tid];
    __syncthreads();
    if (tid < NDBC) {
        const float* wrow = xw + tid * DI;
        float a = 0.f;
        #pragma unroll 8
        for (int k = 0; k < DI; ++k) a += wrow[k] * sx[k];
        sd[tid] = a;
    }
    __syncthreads();
    if (tid < DS) {
        Bm[(size_t)row * DS + tid] = sd[DTR + tid];
        Cm[(size_t)row * DS + tid] = sd[DTR + DS + tid];
    }
    float a = db[tid];
    #pragma unroll
    for (int t = 0; t < DTR; ++t) a += dw[tid * DTR + t] * sd[t];
    dlt[(size_t)row * DI + tid] = (a > 20.f) ? a : log1pf(__expf(a));
}

// ---------------------------------------------------------------------------
// 5) selective scan: grid = B_, block = DI. h[16] in registers. B/C chunks
//    staged straight into LDS with gfx1250 async loads (ASYNCcnt), and the
//    next chunk's delta/x rows prefetched (global_prefetch_b8).
// ---------------------------------------------------------------------------
__global__ void scan_kernel(const float* __restrict__ xc,
                            const float* __restrict__ dlt,
                            const float* __restrict__ Bm,
                            const float* __restrict__ Cm,
                            const float* __restrict__ alog, // (DI, DS)
                            const float* __restrict__ Dp,   // (DI)
                            float* __restrict__ yb) {
    const int b  = blockIdx.x;
    const int di = threadIdx.x;           // 0..255
    float A[DS], h[DS];
    #pragma unroll
    for (int n = 0; n < DS; ++n) {
        A[n] = -__expf(alog[di * DS + n]);
        h[n] = 0.f;
    }
    const float Dv = Dp[di];
    __shared__ float sB[16][DS];
    __shared__ float sC[16][DS];

    for (int l0 = 0; l0 < L_; l0 += 16) {
        __syncthreads();   // previous chunk fully consumed
        // Async-stage 16x16 of B and 16x16 of C (2 KB) into LDS:
        // 128 lanes x 16B each via global_load_async_to_lds_b128.
        if (threadIdx.x < 128) {
            const int sel = threadIdx.x >> 6;              // 0 = B, 1 = C
            const int tt  = threadIdx.x & 63;
            const int li  = tt >> 2;
            const int n4  = (tt & 3) * 4;
            const float* src =
                (sel ? Cm : Bm) + ((size_t)(b * L_ + l0 + li)) * DS + n4;
            unsigned long long ga = (unsigned long long)(uintptr_t)src;
            unsigned lds_off = (unsigned)(uintptr_t)
                (sel ? &sC[li][n4] : &sB[li][n4]);
            asm volatile("global_load_async_to_lds_b128 %0, %1, off"
                         :: "v"(lds_off), "v"(ga) : "memory");
        }
        // prefetch next chunk's delta / x rows while the async copy runs
        if (l0 + 16 < L_) {
            const size_t nrow = (size_t)(b * L_ + l0 + 16) * DI + di;
            __builtin_prefetch(&dlt[nrow], 0, 3);
            __builtin_prefetch(&xc[nrow], 0, 3);
        }
        asm volatile("s_wait_asynccnt 0" ::: "memory");
        __syncthreads();

        for (int li = 0; li < 16; ++li) {
            const size_t row = (size_t)(b * L_ + l0 + li);
            const float d  = dlt[row * DI + di];
            const float xv = xc [row * DI + di];
            const float dx = d * xv;
            float y = Dv * xv;
            #pragma unroll
            for (int n = 0; n < DS; ++n) {
                h[n] = __expf(d * A[n]) * h[n] + dx * sB[li][n];
                y += h[n] * sC[li][n];
            }
            yb[row * DI + di] = y;
        }
    }
}

// ---------------------------------------------------------------------------
// 6) out_proj with fused SiLU gating and residual: out += (y*silu(z)) @ Wo^T
//    Wo:(DM, DI), K = DI = 256. Residual loaded as C matrix. Each wave
//    computes TWO N tiles (n0, n0+64) so the gated A element (one exp per
//    load) feeds two WMMAs. grid = NTOK/16, 128 threads (4 waves).
// ---------------------------------------------------------------------------
__global__ void outproj_wmma_kernel(const float* __restrict__ yb,
                                    const float* __restrict__ xz, // z at [:,DI:]
                                    const float* __restrict__ W,
                                    float* __restrict__ out) {
    const int lane = threadIdx.x & 31;
    const int wave = threadIdx.x >> 5;
    const int m0  = blockIdx.x * 16;
    const int n0a = wave * 16;          // tiles 0..3
    const int n0b = n0a + 64;           // tiles 4..7
    const int t  = lane & 15;
    const int hi = lane >> 4;
    const int ko = hi * 2;

    // preload residual as accumulators (C matrices)
    v8f acca, accb;
    #pragma unroll
    for (int r = 0; r < 8; ++r) {
        const size_t orow = (size_t)(m0 + hi * 8 + r) * DM;
        acca[r] = out[orow + n0a + t];
        accb[r] = out[orow + n0b + t];
    }

    const float* __restrict__ yrow  = yb + (size_t)(m0 + t) * DI;
    const float* __restrict__ zrow  = xz + (size_t)(m0 + t) * DXZ + DI;
    const float* __restrict__ browa = W  + (size_t)(n0a + t) * DI;
    const float* __restrict__ browb = W  + (size_t)(n0b + t) * DI;

    #pragma unroll 4
    for (int k = 0; k < DI; k += 4) {
        v2f a, ba, bb;
        const float z0 = zrow[k + ko];
        const float z1 = zrow[k + ko + 1];
        a[0] = yrow[k + ko]     * z0 * sigmoidf_(z0);
        a[1] = yrow[k + ko + 1] * z1 * sigmoidf_(z1);
        ba[0] = browa[k + ko]; ba[1] = browa[k + ko + 1];
        bb[0] = browb[k + ko]; bb[1] = browb[k + ko + 1];
        acca = __builtin_amdgcn_wmma_f32_16x16x4_f32(
            false, a, false, ba, (short)0, acca, false, false);
        accb = __builtin_amdgcn_wmma_f32_16x16x4_f32(
            false, a, false, bb, (short)0, accb, false, false);
    }
    #pragma unroll
    for (int r = 0; r < 8; ++r) {
        const size_t orow = (size_t)(m0 + hi * 8 + r) * DM;
        out[orow + n0a + t] = acca[r];
        out[orow + n0b + t] = accb[r];
    }
}

// ---------------------------------------------------------------------------
// host launcher
// ---------------------------------------------------------------------------
extern "C" void kernel_launch(void* const* d_in, const int* in_sizes, int n_in,
                              void* d_out, int out_size, void* d_ws, size_t ws_size,
                              hipStream_t stream) {
    const float* x_in   = (const float*)d_in[0];
    const float* norm_w = (const float*)d_in[1];   // (NL, DM)
    const float* in_w   = (const float*)d_in[2];   // (NL, 2*DI, DM)
    const float* conv_w = (const float*)d_in[3];   // (NL, DI, 1, DC)
    const float* conv_b = (const float*)d_in[4];   // (NL, DI)
    const float* xw     = (const float*)d_in[5];   // (NL, 40, DI)
    const float* dw     = (const float*)d_in[6];   // (NL, DI, DTR)
    const float* db     = (const float*)d_in[7];   // (NL, DI)
    const float* alog   = (const float*)d_in[8];   // (NL, DI, DS)
    const float* Dp     = (const float*)d_in[9];   // (NL, DI)
    const float* ow     = (const float*)d_in[10];  // (NL, DM, DI)

    float* out = (float*)d_out;                    // running residual stream

    // workspace carve-up (~94 MB of f32)
    float* ws  = (float*)d_ws;
    float* xn  = ws;                               // NTOK*DM
    float* xz  = xn  + (size_t)NTOK * DM;          // NTOK*512
    float* xc  = xz  + (size_t)NTOK * DXZ;         // NTOK*DI
    float* dlt = xc  + (size_t)NTOK * DI;          // NTOK*DI
    float* Bmb = dlt + (size_t)NTOK * DI;          // NTOK*DS
    float* Cmb = Bmb + (size_t)NTOK * DS;          // NTOK*DS
    float* yb  = Cmb + (size_t)NTOK * DS;          // NTOK*DI

    // out = x (graph-capture-safe D2D copy)
    hipMemcpyAsync(out, x_in, (size_t)NTOK * DM * sizeof(float),
                   hipMemcpyDeviceToDevice, stream);

    for (int i = 0; i < NL; ++i) {
        rmsnorm_kernel<<<NTOK, DM, 0, stream>>>(out, norm_w + (size_t)i * DM, xn);

        inproj_wmma_kernel<<<NTOK / 16, 128, 0, stream>>>(
            xn, in_w + (size_t)i * DXZ * DM, xz);

        conv_silu_kernel<<<(NTOK * DI) / 256, 256, 0, stream>>>(
            xz, conv_w + (size_t)i * DI * DC, conv_b + (size_t)i * DI, xc);

        xproj_kernel<<<NTOK, DI, 0, stream>>>(
            xc, xw + (size_t)i * NDBC * DI, dw + (size_t)i * DI * DTR,
            db + (size_t)i * DI, dlt, Bmb, Cmb);

        scan_kernel<<<B_, DI, 0, stream>>>(
            xc, dlt, Bmb, Cmb, alog + (size_t)i * DI * DS, Dp + (size_t)i * DI, yb);

        outproj_wmma_kernel<<<NTOK / 16, 128, 0, stream>>>(
            yb, xz, ow + (size_t)i * DM * DI, out);
    }
}